// SinkhornDistance_1073741824490
// MI455X (gfx1250) — compile-verified
//
#include <hip/hip_runtime.h>
#include <math.h>

// Sinkhorn on MI455X (gfx1250, wave32).
// C (16x1024x1024 f32 = 64MB) lives in d_out, stays L2-resident (192MB L2),
// and is overwritten in place by pi = exp(M(u,v)) at the end.
// Cost GEMM uses V_WMMA_F32_16X16X4_F32 (exact fp32 tensor path).

#define NB 16
#define P 1024
#define D 64
#define EPSF 0.1f
#define INV_EPS 10.0f
#define MAX_ITER 10

typedef __attribute__((ext_vector_type(2))) float v2f;
typedef __attribute__((ext_vector_type(8))) float v8f;

// ---------------------------------------------------------------------------
// Kernel 1: row squared-norms of x and y, plus v := 0 init (ws is poisoned).
// One thread per row (16384 rows each), float4 loads.
// ---------------------------------------------------------------------------
__global__ void norms_init_kernel(const float* __restrict__ x,
                                  const float* __restrict__ y,
                                  float* __restrict__ xn,
                                  float* __restrict__ yn,
                                  float* __restrict__ v) {
  int r = blockIdx.x * blockDim.x + threadIdx.x;  // 0 .. NB*P-1
  const float4* xr = (const float4*)(x + (size_t)r * D);
  const float4* yr = (const float4*)(y + (size_t)r * D);
  float sx = 0.f, sy = 0.f;
#pragma unroll
  for (int k = 0; k < D / 4; ++k) {
    float4 a = xr[k];
    float4 b = yr[k];
    sx += a.x * a.x + a.y * a.y + a.z * a.z + a.w * a.w;
    sy += b.x * b.x + b.y * b.y + b.z * b.z + b.w * b.w;
  }
  xn[r] = sx;
  yn[r] = sy;
  v[r] = 0.f;
}

// ---------------------------------------------------------------------------
// Kernel 2: C[n,i,j] = xn[i] + yn[j] - 2 * (x . y^T) via fp32 WMMA.
// One wave per 16x16 output tile; 16 x V_WMMA_F32_16X16X4_F32 over K=64.
// A (16x4) layout: lanes 0-15 -> M=lane, K pair {k0,k0+1}; lanes 16-31 -> {k0+2,k0+3}.
// B (4x16) of y^T mirrors it: per lane a contiguous float2 of y[j][k..k+1].
// ---------------------------------------------------------------------------
__global__ void cost_wmma_kernel(const float* __restrict__ x,
                                 const float* __restrict__ y,
                                 const float* __restrict__ xn,
                                 const float* __restrict__ yn,
                                 float* __restrict__ C) {
  const int wave = threadIdx.x >> 5;
  const int lane = threadIdx.x & 31;
  const int half = lane >> 4;   // 0: K pair {0,1}, 1: K pair {2,3}
  const int l16  = lane & 15;

  int tile = blockIdx.x * 8 + wave;        // 0 .. 65535 (exact cover)
  int n    = tile >> 12;                   // / 4096 tiles per batch
  int rem  = tile & 4095;
  int iBase = (rem >> 6) << 4;
  int jBase = (rem & 63) << 4;

  const float* xA = x + ((size_t)n * P + iBase + l16) * D + 2 * half;
  const float* yB = y + ((size_t)n * P + jBase + l16) * D + 2 * half;

  v8f acc = {};
#pragma unroll
  for (int k0 = 0; k0 < D; k0 += 4) {
    v2f a = *(const v2f*)(xA + k0);
    v2f b = *(const v2f*)(yB + k0);
    acc = __builtin_amdgcn_wmma_f32_16x16x4_f32(
        /*neg_a=*/false, a, /*neg_b=*/false, b,
        /*c_mod=*/(short)0, acc, /*reuse_a=*/false, /*reuse_b=*/false);
  }

  // D layout: VGPR r -> row (r + 8*half), col l16.
  int j = jBase + l16;
  float ynj = yn[n * P + j];
  float* Cout = C + ((size_t)n * P + iBase + 8 * half) * P + j;
#pragma unroll
  for (int r = 0; r < 8; ++r) {
    int i = iBase + r + 8 * half;
    Cout[(size_t)r * P] = xn[n * P + i] + ynj - 2.0f * acc[r];
  }
}

// ---------------------------------------------------------------------------
// Kernel 3: u[n,i] = lm - eps * lse_j((v[j] - C[n,i,j]) / eps)
// One 256-thread block per row; 4 elems/thread online lse, LDS tree combine.
// ---------------------------------------------------------------------------
__global__ void u_update_kernel(const float* __restrict__ C,
                                const float* __restrict__ v,
                                float* __restrict__ u, float lm) {
  int row = blockIdx.x;        // n*P + i
  int n = row >> 10;
  const float* Crow = C + (size_t)row * P;
  const float* vn = v + n * P;
  int t = threadIdx.x;

  float m = (vn[t] - Crow[t]) * INV_EPS;
  float s = 1.f;
#pragma unroll
  for (int k = 1; k < P / 256; ++k) {
    int j = t + k * 256;
    float val = (vn[j] - Crow[j]) * INV_EPS;
    float mn = fmaxf(m, val);
    s = s * __expf(m - mn) + __expf(val - mn);
    m = mn;
  }

  __shared__ float sm[256];
  __shared__ float ss[256];
  sm[t] = m;
  ss[t] = s;
  __syncthreads();
  for (int off = 128; off > 0; off >>= 1) {
    if (t < off) {
      float m2 = sm[t + off], s2 = ss[t + off];
      float mn = fmaxf(sm[t], m2);
      ss[t] = ss[t] * __expf(sm[t] - mn) + s2 * __expf(m2 - mn);
      sm[t] = mn;
    }
    __syncthreads();
  }
  if (t == 0) u[row] = lm - EPSF * (sm[0] + __logf(ss[0]));
}

// ---------------------------------------------------------------------------
// Kernel 4: v[n,j] = lm - eps * lse_i((u[i] - C[n,i,j]) / eps)
// Block (32,8): 32 consecutive columns (coalesced), 8-way row striding.
// ---------------------------------------------------------------------------
__global__ void v_update_kernel(const float* __restrict__ C,
                                const float* __restrict__ u,
                                float* __restrict__ v, float lm) {
  int n = blockIdx.y;
  int tx = threadIdx.x;
  int ty = threadIdx.y;
  int j = blockIdx.x * 32 + tx;
  const float* Cn = C + (size_t)n * P * P;
  const float* un = u + n * P;

  float m = (un[ty] - Cn[(size_t)ty * P + j]) * INV_EPS;
  float s = 1.f;
  for (int i = ty + 8; i < P; i += 8) {
    float val = (un[i] - Cn[(size_t)i * P + j]) * INV_EPS;
    float mn = fmaxf(m, val);
    s = s * __expf(m - mn) + __expf(val - mn);
    m = mn;
  }

  __shared__ float sm[8][32];
  __shared__ float ss[8][32];
  sm[ty][tx] = m;
  ss[ty][tx] = s;
  __syncthreads();
  if (ty == 0) {
#pragma unroll
    for (int r = 1; r < 8; ++r) {
      float m2 = sm[r][tx], s2 = ss[r][tx];
      float mn = fmaxf(m, m2);
      s = s * __expf(m - mn) + s2 * __expf(m2 - mn);
      m = mn;
    }
    v[n * P + j] = lm - EPSF * (m + __logf(s));
  }
}

// ---------------------------------------------------------------------------
// Kernel 5: pi = exp((u_i + v_j - C) / eps), in place over d_out (float4).
// ---------------------------------------------------------------------------
__global__ void pi_kernel(float* __restrict__ C,
                          const float* __restrict__ u,
                          const float* __restrict__ v) {
  size_t t = (size_t)blockIdx.x * blockDim.x + threadIdx.x;
  size_t base = t * 4;
  int n = (int)(base >> 20);
  int i = (int)((base >> 10) & (P - 1));
  int j = (int)(base & (P - 1));          // 4 | P, so all 4 in same row
  float ui = u[n * P + i];
  const float* vn = v + n * P;
  float4 c4 = *(float4*)(C + base);
  float4 o;
  o.x = __expf((ui + vn[j + 0] - c4.x) * INV_EPS);
  o.y = __expf((ui + vn[j + 1] - c4.y) * INV_EPS);
  o.z = __expf((ui + vn[j + 2] - c4.z) * INV_EPS);
  o.w = __expf((ui + vn[j + 3] - c4.w) * INV_EPS);
  *(float4*)(C + base) = o;
}

// ---------------------------------------------------------------------------
extern "C" void kernel_launch(void* const* d_in, const int* in_sizes, int n_in,
                              void* d_out, int out_size, void* d_ws, size_t ws_size,
                              hipStream_t stream) {
  const float* y = (const float*)d_in[0];  // setup_inputs order: y first
  const float* x = (const float*)d_in[1];
  float* C = (float*)d_out;                // 16*1024*1024 f32, doubles as pi

  // Workspace: u, v, ||x||^2, ||y||^2 -> 4 * 16384 floats = 256 KB
  float* u  = (float*)d_ws;
  float* v  = u + NB * P;
  float* xn = v + NB * P;
  float* yn = xn + NB * P;

  // eps * log(1/P + 1e-8), identical for mu and nu
  float lm = EPSF * logf(1.0f / (float)P + 1e-8f);

  norms_init_kernel<<<NB * P / 256, 256, 0, stream>>>(x, y, xn, yn, v);

  cost_wmma_kernel<<<NB * (P / 16) * (P / 16) / 8, 256, 0, stream>>>(x, y, xn, yn, C);

  for (int it = 0; it < MAX_ITER; ++it) {
    u_update_kernel<<<NB * P, 256, 0, stream>>>(C, v, u, lm);
    v_update_kernel<<<dim3(P / 32, NB), dim3(32, 8), 0, stream>>>(C, u, v, lm);
  }

  pi_kernel<<<(NB * P * P) / (256 * 4), 256, 0, stream>>>(C, u, v);
}